// SGConvolution_65807488909795
// MI455X (gfx1250) — compile-verified
//
#include <hip/hip_runtime.h>

#define D_FEAT 32
#define TILE 32               // edges per tile (== wave32 lanes)
#define WAVES_PER_BLOCK 8
#define BLOCK_SIZE 256
#define TILES_PER_WAVE 4      // grid-stride depth; enables double-buffer pipeline

#if defined(__has_builtin)
#if __has_builtin(__builtin_amdgcn_global_load_async_to_lds_b32)
#define HAVE_ASYNC_LDS 1
#endif
#if __has_builtin(__builtin_amdgcn_s_wait_asynccnt)
#define HAVE_WAIT_ASYNC 1
#endif
#endif

// 16B-aligned per-edge metadata slot in LDS: {col, row, w, pad}
// -> one ds_load_b128 broadcast per edge in the inner loop.
struct __align__(16) EdgeMeta { int c; int r; float w; int pad; };

// Issue 3 coalesced async copies (ASYNCcnt) of one tile's edge metadata
// into this wave's private LDS staging buffer (packed layout).
__device__ __forceinline__ void sgc_stage_tile(const int*   rows,
                                               const int*   cols,
                                               const float* wgt,
                                               int tile, int n_edges, int lane,
                                               EdgeMeta* dst) {
    int e = tile * TILE + lane;
    if (e > n_edges - 1) e = n_edges - 1;          // clamp tail (dup harmless)
    int* slot = (int*)&dst[lane];
#if HAVE_ASYNC_LDS
    __builtin_amdgcn_global_load_async_to_lds_b32((int*)(cols + e), slot + 0, 0, 0);
    __builtin_amdgcn_global_load_async_to_lds_b32((int*)(rows + e), slot + 1, 0, 0);
    __builtin_amdgcn_global_load_async_to_lds_b32((int*)(wgt + e),  slot + 2, 0, 0);
#else
    slot[0] = cols[e];
    slot[1] = rows[e];
    slot[2] = __float_as_int(wgt[e]);
#endif
}

// One wave owns TILES_PER_WAVE tiles (round-robin, stride = total waves).
// Double-buffered pipeline: async-stage tile t+1, s_wait_asynccnt 3 (in-order
// completion => tile t's 3 copies landed), process tile t while t+1 streams.
__global__ void __launch_bounds__(BLOCK_SIZE)
sgc_spmm_scatter(const float* __restrict__ x,
                 const int*   __restrict__ rows,
                 const int*   __restrict__ cols,
                 const float* __restrict__ wgt,
                 float*       __restrict__ out,
                 int n_edges) {
    __shared__ EdgeMeta s_meta[WAVES_PER_BLOCK][2][TILE];

    const int lane = threadIdx.x & 31;
    const int wv   = threadIdx.x >> 5;
    const int n_waves = gridDim.x * WAVES_PER_BLOCK;
    const int wave_id = blockIdx.x * WAVES_PER_BLOCK + wv;
    const int n_tiles = (n_edges + TILE - 1) / TILE;

    int tile = wave_id;
    if (tile >= n_tiles) return;

    // Prologue: stage first tile into buffer 0.
    sgc_stage_tile(rows, cols, wgt, tile, n_edges, lane, s_meta[wv][0]);
    int cur = 0;

    for (; tile < n_tiles; tile += n_waves) {
        const int next = tile + n_waves;
        const bool have_next = next < n_tiles;

        if (have_next) {
            // Stage tile t+1 into the other buffer while t is consumed.
            sgc_stage_tile(rows, cols, wgt, next, n_edges, lane,
                           s_meta[wv][cur ^ 1]);
#if HAVE_WAIT_ASYNC
            __builtin_amdgcn_s_wait_asynccnt(3);   // current tile's 3 copies done
#elif HAVE_ASYNC_LDS
            asm volatile("s_wait_asynccnt 3" ::: "memory");
#endif
        } else {
#if HAVE_WAIT_ASYNC
            __builtin_amdgcn_s_wait_asynccnt(0);
#elif HAVE_ASYNC_LDS
            asm volatile("s_wait_asynccnt 0" ::: "memory");
#endif
        }
        asm volatile("" ::: "memory");             // keep LDS reads below waits

        // Each lane prefetches the x-row of *its* edge in the current tile:
        // 32 scattered 128B rows go in flight before the serial loop needs them.
        {
            const char* prow = (const char*)(x + s_meta[wv][cur][lane].c * D_FEAT);
            __builtin_prefetch(prow, 0, 3);
            __builtin_prefetch(prow + 64, 0, 3);
        }

        const int base = tile * TILE;
        int cnt = n_edges - base;
        if (cnt > TILE) cnt = TILE;

        if (cnt == TILE) {
            #pragma unroll 8
            for (int j = 0; j < TILE; ++j) {
                const int4 m = *(const int4*)&s_meta[wv][cur][j]; // 1x ds_load_b128
                const float w = __int_as_float(m.z);
                const float v = x[m.x * D_FEAT + lane] * w;       // 128B coalesced
                atomicAdd(out + m.y * D_FEAT + lane, v);          // L2-side atomic
            }
        } else {
            for (int j = 0; j < cnt; ++j) {
                const int4 m = *(const int4*)&s_meta[wv][cur][j];
                const float w = __int_as_float(m.z);
                const float v = x[m.x * D_FEAT + lane] * w;
                atomicAdd(out + m.y * D_FEAT + lane, v);
            }
        }
        cur ^= 1;
    }
}

__global__ void __launch_bounds__(BLOCK_SIZE)
sgc_zero_kernel(float4* __restrict__ p, int n4) {
    int i = blockIdx.x * blockDim.x + threadIdx.x;
    if (i < n4) p[i] = make_float4(0.f, 0.f, 0.f, 0.f);
}

extern "C" void kernel_launch(void* const* d_in, const int* in_sizes, int n_in,
                              void* d_out, int out_size, void* d_ws, size_t ws_size,
                              hipStream_t stream) {
    const float* x       = (const float*)d_in[0];
    const int*   adj_row = (const int*)  d_in[1];   // JAX default x64-off: int32
    const int*   adj_col = (const int*)  d_in[2];
    const float* adj_w   = (const float*)d_in[3];

    const int n_edges = in_sizes[1];
    const int n_vals  = in_sizes[0];                // N_NODES * D_FEAT
    float* tmp = (float*)d_ws;                      // hop-1 result (12.8 MB)
    float* out = (float*)d_out;

    const int n4      = n_vals / 4;
    const int zblocks = (n4 + BLOCK_SIZE - 1) / BLOCK_SIZE;
    const int n_tiles = (n_edges + TILE - 1) / TILE;
    const int n_waves = (n_tiles + TILES_PER_WAVE - 1) / TILES_PER_WAVE;
    int sblocks = (n_waves + WAVES_PER_BLOCK - 1) / WAVES_PER_BLOCK;
    if (sblocks < 1) sblocks = 1;

    // hop 1: tmp = A @ x
    sgc_zero_kernel<<<zblocks, BLOCK_SIZE, 0, stream>>>((float4*)tmp, n4);
    sgc_spmm_scatter<<<sblocks, BLOCK_SIZE, 0, stream>>>(x, adj_row, adj_col,
                                                         adj_w, tmp, n_edges);
    // hop 2: out = A @ tmp
    sgc_zero_kernel<<<zblocks, BLOCK_SIZE, 0, stream>>>((float4*)out, n4);
    sgc_spmm_scatter<<<sblocks, BLOCK_SIZE, 0, stream>>>(tmp, adj_row, adj_col,
                                                         adj_w, out, n_edges);
}